// PackedINTSymmetricWeightsDecompressor_74440373174409
// MI455X (gfx1250) — compile-verified
//
#include <hip/hip_runtime.h>

// PackedINTSymmetricWeightsDecompressor for gfx1250 (MI455X).
//
// Pure streaming dequant: 128 MB int32-packed nibbles + 2 MB scales in,
// 256 MB fp32 out. HBM-bound (~16.6 us floor at 23.3 TB/s). No matrix ops
// exist in this workload, so no WMMA; the win is 128-bit coalesced traffic,
// non-temporal cache hints (footprint > 192 MB L2), and global_prefetch_b8
// run-ahead on the weight stream.

typedef __attribute__((ext_vector_type(4))) float f32x4;
typedef __attribute__((ext_vector_type(4))) int   i32x4;

#define TPB 256
#define PACKED_PER_THREAD 8   // 8 int32 "bytes" -> 16 fp32 outputs per thread

__global__ __launch_bounds__(TPB) void
PackedINTSymmetricWeightsDecompressor_kernel(const int*   __restrict__ weight,
                                             const float* __restrict__ scale,
                                             float*       __restrict__ out,
                                             int n_packed)
{
    const int t  = blockIdx.x * TPB + threadIdx.x;
    const int p0 = t * PACKED_PER_THREAD;           // first packed element
    if (p0 >= n_packed) return;

    // Run-ahead prefetch (~1 MB ahead in the weight stream).
    // Lowers to global_prefetch_b8 on gfx1250 (probe-confirmed).
    const int AHEAD = (1 << 20) / (int)sizeof(int); // elements
    if (p0 + AHEAD < n_packed)
        __builtin_prefetch(weight + p0 + AHEAD, /*rw=*/0, /*locality=*/1);

    // GROUP_SIZE/2 = 64 packed elements per group; 8 | 64, so one scale/thread.
    const float s = scale[t >> 3];

    // Two 128-bit non-temporal loads: 8 packed bytes (stored one per int32).
    const i32x4* wp = (const i32x4*)(weight + p0);
    const i32x4  w0 = __builtin_nontemporal_load(wp);
    const i32x4  w1 = __builtin_nontemporal_load(wp + 1);

    const int b[8] = { w0.x, w0.y, w0.z, w0.w, w1.x, w1.y, w1.z, w1.w };

    // 16 outputs: out[2p] = (low-8)*s, out[2p+1] = (high-8)*s.
    f32x4* op = (f32x4*)(out + (size_t)p0 * 2);
#pragma unroll
    for (int i = 0; i < 4; ++i) {
        const int a = b[2 * i];
        const int c = b[2 * i + 1];
        f32x4 v;
        v.x = (float)(( a        & 0xF) - 8) * s;
        v.y = (float)(((a >> 4)  & 0xF) - 8) * s;
        v.z = (float)(( c        & 0xF) - 8) * s;
        v.w = (float)(((c >> 4)  & 0xF) - 8) * s;
        __builtin_nontemporal_store(v, op + i);   // 128-bit NT store
    }
}

extern "C" void kernel_launch(void* const* d_in, const int* in_sizes, int n_in,
                              void* d_out, int out_size, void* d_ws, size_t ws_size,
                              hipStream_t stream)
{
    (void)n_in; (void)out_size; (void)d_ws; (void)ws_size;

    const int*   weight = (const int*)  d_in[0]; // (8192, 64, 64) int32, one packed byte each
    const float* scale  = (const float*)d_in[1]; // (8192, 64) fp32
    float*       out    = (float*)      d_out;   // (8192, 8192) fp32

    const int n_packed = in_sizes[0];            // 33,554,432
    const int threads  = (n_packed + PACKED_PER_THREAD - 1) / PACKED_PER_THREAD;
    const int blocks   = (threads + TPB - 1) / TPB;

    PackedINTSymmetricWeightsDecompressor_kernel<<<blocks, TPB, 0, stream>>>(
        weight, scale, out, n_packed);
}